// GraphConv_85968065397178
// MI455X (gfx1250) — compile-verified
//
#include <hip/hip_runtime.h>
#include <hip/hip_bf16.h>

#define N_USERS    50000
#define N_ENTITIES 100000
#define N_RELATIONS 32
#define N_FACTORS  8
#define D          64
#define N_KG       1500000
#define N_INTER    1000000

typedef float v2f __attribute__((ext_vector_type(2)));
typedef float v8f __attribute__((ext_vector_type(8)));

__device__ __forceinline__ void atomic_add_f32(float* p, float v) {
    __hip_atomic_fetch_add(p, v, __ATOMIC_RELAXED, __HIP_MEMORY_SCOPE_AGENT);
}

// ---------------------------------------------------------------- utilities

__global__ void fill0_kernel(float* __restrict__ p, int n, const int* __restrict__ nh, int hop) {
    if (hop >= *nh) return;
    int i = blockIdx.x * blockDim.x + threadIdx.x;
    if (i < n) p[i] = 0.0f;
}

__global__ void init4_kernel(const float* __restrict__ src,
                             float* __restrict__ a, float* __restrict__ b,
                             float* __restrict__ c, float* __restrict__ d, int n) {
    int i = blockIdx.x * blockDim.x + threadIdx.x;
    if (i < n) {
        float v = src[i];
        a[i] = v; b[i] = v; c[i] = v; d[i] = v;
    }
}

// relation_intent_emb[8x64] = proj[8x32] @ rel[32x64]
__global__ void rel_intent_kernel(const float* __restrict__ proj,
                                  const float* __restrict__ rel,
                                  float* __restrict__ out) {
    int i = blockIdx.x * blockDim.x + threadIdx.x;
    if (i >= N_FACTORS * D) return;
    int f = i >> 6, d = i & 63;
    float s = 0.0f;
#pragma unroll
    for (int r = 0; r < N_RELATIONS; ++r)
        s += proj[f * N_RELATIONS + r] * rel[r * D + d];
    out[i] = s;
}

// ---------------------------------------------------------------- scatters

// user_agg[iu[e]] += vals[e] * ent[ii[e]]   (16 threads/edge, float4 each)
__global__ void scatter_user_agg(const float* __restrict__ ent,
                                 const float* __restrict__ vals,
                                 const int* __restrict__ iu, const int* __restrict__ ii,
                                 float* __restrict__ uagg,
                                 const int* __restrict__ nh, int hop) {
    if (hop >= *nh) return;
    long idx = (long)blockIdx.x * blockDim.x + threadIdx.x;
    long e = idx >> 4;
    int  q = (int)(idx & 15);
    if (e >= N_INTER) return;
    float w = vals[e];
    const float4 v = *(const float4*)(ent + (long)ii[e] * D + q * 4);
    float* dst = uagg + (long)iu[e] * D + q * 4;
    atomic_add_f32(dst + 0, w * v.x);
    atomic_add_f32(dst + 1, w * v.y);
    atomic_add_f32(dst + 2, w * v.z);
    atomic_add_f32(dst + 3, w * v.w);
}

// ent_agg[ii[e]] += usr[iu[e]] ; cnt[ii[e]] += 1
__global__ void scatter_ent_hist(const float* __restrict__ usr,
                                 const int* __restrict__ iu, const int* __restrict__ ii,
                                 float* __restrict__ eagg, float* __restrict__ ecnt,
                                 const int* __restrict__ nh, int hop) {
    if (hop >= *nh) return;
    long idx = (long)blockIdx.x * blockDim.x + threadIdx.x;
    long e = idx >> 4;
    int  q = (int)(idx & 15);
    if (e >= N_INTER) return;
    const float4 v = *(const float4*)(usr + (long)iu[e] * D + q * 4);
    int h = ii[e];
    float* dst = eagg + (long)h * D + q * 4;
    atomic_add_f32(dst + 0, v.x);
    atomic_add_f32(dst + 1, v.y);
    atomic_add_f32(dst + 2, v.z);
    atomic_add_f32(dst + 3, v.w);
    if (q == 0) atomic_add_f32(ecnt + h, 1.0f);
}

// ent_agg[head[e]] += ent[tail[e]] * rel_emb[type[e]] ; cnt[head[e]] += 1
__global__ void scatter_ent_rel(const float* __restrict__ ent,
                                const float* __restrict__ rel_emb,
                                const int* __restrict__ eh, const int* __restrict__ et,
                                const int* __restrict__ etype,
                                float* __restrict__ eagg, float* __restrict__ ecnt,
                                const int* __restrict__ nh, int hop) {
    if (hop >= *nh) return;
    long idx = (long)blockIdx.x * blockDim.x + threadIdx.x;
    long e = idx >> 4;
    int  q = (int)(idx & 15);
    if (e >= N_KG) return;
    const float4 v = *(const float4*)(ent + (long)et[e] * D + q * 4);
    const float4 r = *(const float4*)(rel_emb + (long)etype[e] * D + q * 4);
    int h = eh[e];
    float* dst = eagg + (long)h * D + q * 4;
    atomic_add_f32(dst + 0, v.x * r.x);
    atomic_add_f32(dst + 1, v.y * r.y);
    atomic_add_f32(dst + 2, v.z * r.z);
    atomic_add_f32(dst + 3, v.w * r.w);
    if (q == 0) atomic_add_f32(ecnt + h, 1.0f);
}

// mean + l2norm -> cur ; res += cur.  One wave32 per entity row, 2 floats/lane.
__global__ void finalize_entity(const float* __restrict__ eagg, const float* __restrict__ ecnt,
                                float* __restrict__ cur, float* __restrict__ res,
                                const int* __restrict__ nh, int hop) {
    if (hop >= *nh) return;
    int gid = blockIdx.x * blockDim.x + threadIdx.x;
    int ent = gid >> 5;
    int lane = gid & 31;
    if (ent >= N_ENTITIES) return;
    float c = fmaxf(ecnt[ent], 1.0f);
    float2 a = *(const float2*)(eagg + (long)ent * D + lane * 2);
    a.x /= c; a.y /= c;
    float ss = a.x * a.x + a.y * a.y;
    ss += __shfl_xor(ss, 1, 32);
    ss += __shfl_xor(ss, 2, 32);
    ss += __shfl_xor(ss, 4, 32);
    ss += __shfl_xor(ss, 8, 32);
    ss += __shfl_xor(ss, 16, 32);
    float inv = 1.0f / fmaxf(sqrtf(ss), 1e-12f);
    a.x *= inv; a.y *= inv;
    *(float2*)(cur + (long)ent * D + lane * 2) = a;
    float2* rp = (float2*)(res + (long)ent * D + lane * 2);
    float2 r0 = *rp;
    r0.x += a.x; r0.y += a.y;
    *rp = r0;
}

// ------------------------------------------------- intent_user (WMMA f32 16x16x4)
// score = softmax(U @ W^T); out = l2norm((score @ W) * agg + agg); res += out
// One wave32 handles a 16-user tile. All operand loads are branch-free:
//  - A rows clamped (garbage rows only affect their own, never-stored outputs)
//  - B zero-padding baked into LDS data (s_intent rows 8..15 = 0)
//  - paired K values contiguous -> b64 loads

__global__ __launch_bounds__(256)
void intent_user_wmma(float* __restrict__ usr, float* __restrict__ usr_res,
                      const float* __restrict__ uagg, const float* __restrict__ intent,
                      const int* __restrict__ nh, int hop) {
    if (hop >= *nh) return;

    __shared__ float s_intent[16][D];     // W padded to 16 rows (rows >= 8 are zero)
    __shared__ float s_w2[D][N_FACTORS];  // W transposed: s_w2[n][k] = W[k][n]
    __shared__ float s_scores[8][16][8];  // per-wave score tile

    for (int i = threadIdx.x; i < 16 * D; i += blockDim.x)
        ((float*)s_intent)[i] = (i < N_FACTORS * D) ? intent[i] : 0.0f;
    for (int i = threadIdx.x; i < D * N_FACTORS; i += blockDim.x) {
        int n = i >> 3, k = i & 7;
        s_w2[n][k] = intent[k * D + n];
    }
    __syncthreads();

    const int lane = threadIdx.x & 31;
    const int wid  = threadIdx.x >> 5;
    const int col  = lane & 15;          // A-row index / B-col index
    const int hi   = lane >> 4;          // half-wave -> K offset +2
    const int tileBase = (blockIdx.x * 8 + wid) * 16;

    // ---- GEMM1: logits = U(16x64) @ W^T(64x16, cols>=8 zero-padded) ----
    v8f c = {0.0f, 0.0f, 0.0f, 0.0f, 0.0f, 0.0f, 0.0f, 0.0f};
    const int  m  = tileBase + col;
    const bool mv = (m < N_USERS);
    const float* urow = usr + (size_t)(mv ? m : 0) * D;   // clamped, branch-free
#pragma unroll
    for (int k0 = 0; k0 < D; k0 += 4) {
        const int ka = k0 + 2 * hi;
        v2f a = *(const v2f*)(urow + ka);                 // global_load_b64
        v2f b = *(const v2f*)&s_intent[col][ka];          // ds_load_b64
        c = __builtin_amdgcn_wmma_f32_16x16x4_f32(false, a, false, b, (short)0, c, false, false);
    }

    // ---- softmax over the 8 factor columns (lanes 0-7 / 16-23 hold them) ----
    float p[8];
#pragma unroll
    for (int j = 0; j < 8; ++j) {
        float v = c[j];
        float mx = v;
        mx = fmaxf(mx, __shfl_xor(mx, 1, 8));
        mx = fmaxf(mx, __shfl_xor(mx, 2, 8));
        mx = fmaxf(mx, __shfl_xor(mx, 4, 8));
        float e = __expf(v - mx);
        float s = e;
        s += __shfl_xor(s, 1, 8);
        s += __shfl_xor(s, 2, 8);
        s += __shfl_xor(s, 4, 8);
        p[j] = e / s;
    }
    if (col < N_FACTORS) {
#pragma unroll
        for (int j = 0; j < 8; ++j)
            s_scores[wid][j + 8 * hi][col] = p[j];
    }
    __syncthreads();

    // ---- GEMM2: T = score(16x8) @ W(8x64), N in 4 tiles of 16 ----
    v8f acc[4];
#pragma unroll
    for (int t = 0; t < 4; ++t)
        acc[t] = (v8f){0.0f, 0.0f, 0.0f, 0.0f, 0.0f, 0.0f, 0.0f, 0.0f};
#pragma unroll
    for (int ks = 0; ks < N_FACTORS; ks += 4) {
        const int ka = ks + 2 * hi;
        v2f a = *(const v2f*)&s_scores[wid][col][ka];     // ds_load_b64
#pragma unroll
        for (int t = 0; t < 4; ++t) {
            v2f b = *(const v2f*)&s_w2[t * 16 + col][ka]; // ds_load_b64
            acc[t] = __builtin_amdgcn_wmma_f32_16x16x4_f32(false, a, false, b, (short)0, acc[t], false, false);
        }
    }

    // ---- epilogue: out = (T + 1) * user_agg ; l2norm per row; res += ----
#pragma unroll
    for (int j = 0; j < 8; ++j) {
        const int  r  = tileBase + j + 8 * hi;
        const bool rv = (r < N_USERS);
        const float* ua = uagg + (size_t)(rv ? r : 0) * D;  // clamped
        float s2 = 0.0f;
#pragma unroll
        for (int t = 0; t < 4; ++t) {
            float val = (acc[t][j] + 1.0f) * ua[t * 16 + col];
            acc[t][j] = val;
            s2 += val * val;
        }
        s2 += __shfl_xor(s2, 1, 16);
        s2 += __shfl_xor(s2, 2, 16);
        s2 += __shfl_xor(s2, 4, 16);
        s2 += __shfl_xor(s2, 8, 16);
        float inv = 1.0f / fmaxf(sqrtf(s2), 1e-12f);
        if (rv) {
#pragma unroll
            for (int t = 0; t < 4; ++t) {
                float nv = acc[t][j] * inv;
                usr[(size_t)r * D + t * 16 + col] = nv;
                usr_res[(size_t)r * D + t * 16 + col] += nv;
            }
        }
    }
}

// ---------------------------------------------------------------- concat

__global__ void concat2_kernel(const float* __restrict__ a, const float* __restrict__ b,
                               float* __restrict__ out, int rows) {
    int idx = blockIdx.x * blockDim.x + threadIdx.x;
    if (idx >= rows * D) return;
    int i = idx >> 6, d = idx & 63;
    out[(size_t)i * 128 + d]      = a[idx];
    out[(size_t)i * 128 + 64 + d] = b[idx];
}

// ---------------------------------------------------------------- launch

extern "C" void kernel_launch(void* const* d_in, const int* in_sizes, int n_in,
                              void* d_out, int out_size, void* d_ws, size_t ws_size,
                              hipStream_t stream) {
    const float* user_emb    = (const float*)d_in[0];
    const float* entity_emb  = (const float*)d_in[1];
    const float* rel_emb     = (const float*)d_in[2];
    const float* proj        = (const float*)d_in[3];
    const float* hist_intent = (const float*)d_in[4];
    const float* int_vals    = (const float*)d_in[5];
    const int*   edge_head   = (const int*)d_in[6];
    const int*   edge_tail   = (const int*)d_in[7];
    const int*   edge_type   = (const int*)d_in[8];
    const int*   iu          = (const int*)d_in[9];
    const int*   ii          = (const int*)d_in[10];
    const int*   nhops       = (const int*)d_in[11];

    const size_t ENT = (size_t)N_ENTITIES * D;   // 6.4M
    const size_t USR = (size_t)N_USERS * D;      // 3.2M

    // d_out layout: entity_res[NE*128] | user_res[NU*128] | h_ent_res | r_ent_res | h_usr_res | r_usr_res
    float* out        = (float*)d_out;
    float* entity_res = out;
    float* user_res   = out + (size_t)N_ENTITIES * 128;
    float* h_ent_res  = user_res + (size_t)N_USERS * 128;
    float* r_ent_res  = h_ent_res + ENT;
    float* h_usr_res  = r_ent_res + ENT;
    float* r_usr_res  = h_usr_res + USR;

    // "current" embeddings live in the (not-yet-needed) concat regions of d_out
    float* h_ent = entity_res;            // 6.4M
    float* r_ent = entity_res + ENT;      // 6.4M  (entity_res region = 12.8M total)
    float* h_usr = user_res;              // 3.2M
    float* r_usr = user_res + USR;        // 3.2M  (user_res region = 6.4M total)

    // workspace: user_agg | ent_agg | ent_cnt | rel_intent_emb
    float* ws   = (float*)d_ws;
    float* uagg = ws;
    float* eagg = uagg + USR;
    float* ecnt = eagg + ENT;
    float* rint = ecnt + N_ENTITIES;

    const int B = 256;
    const int gENT   = (int)((ENT + B - 1) / B);
    const int gUSR   = (int)((USR + B - 1) / B);
    const int gEAGG  = (int)((ENT + N_ENTITIES + B - 1) / B);
    const int gINTER = (int)(((long)N_INTER * 16 + B - 1) / B);
    const int gKG    = (int)(((long)N_KG * 16 + B - 1) / B);
    const int gFIN   = (int)(((long)N_ENTITIES * 32 + B - 1) / B);
    const int gIUS   = (N_USERS + 127) / 128;   // 8 waves * 16 users per block

    init4_kernel<<<gENT, B, 0, stream>>>(entity_emb, h_ent, r_ent, h_ent_res, r_ent_res, (int)ENT);
    init4_kernel<<<gUSR, B, 0, stream>>>(user_emb, h_usr, r_usr, h_usr_res, r_usr_res, (int)USR);
    rel_intent_kernel<<<2, B, 0, stream>>>(proj, rel_emb, rint);

    for (int hop = 0; hop < 2; ++hop) {
        // ---- history-intent branch ----
        fill0_kernel<<<gUSR, B, 0, stream>>>(uagg, (int)USR, nhops, hop);
        scatter_user_agg<<<gINTER, B, 0, stream>>>(h_ent, int_vals, iu, ii, uagg, nhops, hop);
        fill0_kernel<<<gEAGG, B, 0, stream>>>(eagg, (int)(ENT + N_ENTITIES), nhops, hop);
        scatter_ent_hist<<<gINTER, B, 0, stream>>>(h_usr, iu, ii, eagg, ecnt, nhops, hop);
        finalize_entity<<<gFIN, B, 0, stream>>>(eagg, ecnt, h_ent, h_ent_res, nhops, hop);
        intent_user_wmma<<<gIUS, B, 0, stream>>>(h_usr, h_usr_res, uagg, hist_intent, nhops, hop);

        // ---- relation-intent branch ----
        fill0_kernel<<<gUSR, B, 0, stream>>>(uagg, (int)USR, nhops, hop);
        scatter_user_agg<<<gINTER, B, 0, stream>>>(r_ent, int_vals, iu, ii, uagg, nhops, hop);
        fill0_kernel<<<gEAGG, B, 0, stream>>>(eagg, (int)(ENT + N_ENTITIES), nhops, hop);
        scatter_ent_rel<<<gKG, B, 0, stream>>>(r_ent, rel_emb, edge_head, edge_tail, edge_type,
                                               eagg, ecnt, nhops, hop);
        finalize_entity<<<gFIN, B, 0, stream>>>(eagg, ecnt, r_ent, r_ent_res, nhops, hop);
        intent_user_wmma<<<gIUS, B, 0, stream>>>(r_usr, r_usr_res, uagg, rint, nhops, hop);
    }

    // final concatenation (reads res regions, overwrites the cur-embedding regions)
    concat2_kernel<<<gENT, B, 0, stream>>>(h_ent_res, r_ent_res, entity_res, N_ENTITIES);
    concat2_kernel<<<gUSR, B, 0, stream>>>(h_usr_res, r_usr_res, user_res, N_USERS);
}